// InstanceDiscLoss_22728966930460
// MI455X (gfx1250) — compile-verified
//
#include <hip/hip_runtime.h>
#include <hip/hip_bf16.h>

typedef __attribute__((ext_vector_type(2))) float v2f;
typedef __attribute__((ext_vector_type(8))) float v8f;

#define NB   2
#define CCH  32
#define NVOX (32*128*128)
#define BQ   8
#define INV_TEMP (1.0f/0.3f)
#define WEIGHT_  1.0f

// workspace float layout
#define WS_SUMS 0                         // NB*BQ*CCH = 512
#define WS_CNT  (WS_SUMS + NB*BQ*CCH)     // NB
#define WS_TN   (WS_CNT + NB)             // NB*BQ*CCH = 512
#define WS_ACC  (WS_TN + NB*BQ*CCH)       // NB
#define WS_TOTAL (WS_ACC + NB)

__global__ void k_zero(float* ws, int n) {
    int i = blockIdx.x * blockDim.x + threadIdx.x;
    if (i < n) ws[i] = 0.0f;
}

// Pass 1: masked per-instance feature sums + labeled-voxel count (LDS staged).
__global__ void k_sums(const float* __restrict__ ft, const int* __restrict__ tm,
                       float* __restrict__ ws) {
    __shared__ float s[BQ * CCH];
    __shared__ float scnt;
    const int b   = blockIdx.y;
    const int tid = threadIdx.x;
    for (int i = tid; i < BQ * CCH; i += blockDim.x) s[i] = 0.0f;
    if (tid == 0) scnt = 0.0f;
    __syncthreads();

    const float* f = ft + (size_t)b * CCH * NVOX;
    const int*   t = tm + (size_t)b * NVOX;
    const int stride = gridDim.x * blockDim.x;
    int cnt = 0;
    for (int v = blockIdx.x * blockDim.x + tid; v < NVOX; v += stride) {
        const int lbl = t[v];
        if (lbl >= 1 && lbl <= BQ) {
            cnt++;
            float* row = s + (lbl - 1) * CCH;
            #pragma unroll
            for (int c = 0; c < CCH; ++c)
                atomicAdd(&row[c], f[(size_t)c * NVOX + v]);
        }
    }
    atomicAdd(&scnt, (float)cnt);
    __syncthreads();
    for (int i = tid; i < BQ * CCH; i += blockDim.x)
        atomicAdd(&ws[WS_SUMS + b * BQ * CCH + i], s[i]);
    if (tid == 0) atomicAdd(&ws[WS_CNT + b], scnt);
}

// Pass 2: t = sums/count, L2-normalize rows.  One block, 512 threads.
__global__ void k_norm(float* __restrict__ ws) {
    __shared__ float t[NB * BQ * CCH];
    const int tid = threadIdx.x;               // 0..511
    const int b   = tid / (BQ * CCH);
    const float cnt = ws[WS_CNT + b];
    const float val = ws[WS_SUMS + tid] / cnt;
    t[tid] = val;
    __syncthreads();
    const int row = tid / CCH;                 // b*BQ + q
    float ss = 0.0f;
    #pragma unroll
    for (int c = 0; c < CCH; ++c) { const float x = t[row * CCH + c]; ss += x * x; }
    const float nrm = fmaxf(sqrtf(ss), 1e-12f);
    ws[WS_TN + tid] = val / nrm;
}

// Pass 3: logits via V_WMMA_F32_16X16X4_F32, per-voxel logsumexp - own logit.
// A = tn (M=q rows, rows 8..15 zero-padded), B = 16 ft columns, K=32 in 8 steps.
// D layout: lane L<16 holds q=0..7 for column (col0+L) in acc[0..7].
__global__ void k_loss(const float* __restrict__ ft, const int* __restrict__ tm,
                       const float* __restrict__ ws_tn, float* __restrict__ ws_acc) {
    const int b    = blockIdx.y;
    const int lane = threadIdx.x & 31;
    const int wave = threadIdx.x >> 5;
    const int col0 = (blockIdx.x * 8 + wave) * 16;

    const float* f  = ft + (size_t)b * CCH * NVOX;
    const float* tn = ws_tn + b * BQ * CCH;    // 8x32 row-major

    // A fragments per ISA 32-bit A 16x4 layout:
    //   VGPR0: K=k0   (lanes 0-15) / K=k0+2 (lanes 16-31)
    //   VGPR1: K=k0+1 (lanes 0-15) / K=k0+3 (lanes 16-31)
    const int m    = lane & 15;
    const int koff = (lane >> 4) << 1;         // 0 or 2
    v2f a[8];
    #pragma unroll
    for (int ks = 0; ks < 8; ++ks) {
        const int k0 = ks * 4 + koff;
        a[ks].x = (m < BQ) ? tn[m * CCH + k0]     : 0.0f;
        a[ks].y = (m < BQ) ? tn[m * CCH + k0 + 1] : 0.0f;
    }

    const int coln = col0 + (lane & 15);       // B column this lane feeds
    v8f acc = {};
    #pragma unroll
    for (int ks = 0; ks < 8; ++ks) {
        const int kb = ks * 4 + koff;
        v2f bm;
        bm.x = f[(size_t)kb       * NVOX + coln];
        bm.y = f[(size_t)(kb + 1) * NVOX + coln];
        // D = A x B + C   (full-EXEC here; divergence only after the matmul)
        acc = __builtin_amdgcn_wmma_f32_16x16x4_f32(
                  false, a[ks], false, bm, (short)0, acc, false, false);
    }

    float contrib = 0.0f;
    if (lane < 16) {
        float l[BQ];
        float mx = -3.402823466e38f;
        #pragma unroll
        for (int q = 0; q < BQ; ++q) { l[q] = acc[q] * INV_TEMP; mx = fmaxf(mx, l[q]); }
        float se = 0.0f;
        #pragma unroll
        for (int q = 0; q < BQ; ++q) se += __expf(l[q] - mx);
        const float lse = mx + __logf(se);
        const int lbl = tm[(size_t)b * NVOX + coln];
        if (lbl >= 1 && lbl <= BQ) contrib = lse - l[lbl - 1];
    }
    #pragma unroll
    for (int off = 16; off >= 1; off >>= 1)
        contrib += __shfl_xor(contrib, off, 32);
    if (lane == 0) atomicAdd(&ws_acc[b], contrib);
}

__global__ void k_final(const float* __restrict__ ws, float* __restrict__ out) {
    float s = 0.0f;
    #pragma unroll
    for (int b = 0; b < NB; ++b)
        s += ws[WS_ACC + b] / ws[WS_CNT + b];
    out[0] = (s / (float)NB) * WEIGHT_;
}

extern "C" void kernel_launch(void* const* d_in, const int* in_sizes, int n_in,
                              void* d_out, int out_size, void* d_ws, size_t ws_size,
                              hipStream_t stream) {
    const float* ft = (const float*)d_in[0];   // [B, C, D, H, W] f32
    const int*   tm = (const int*)d_in[1];     // [B, 1, D, H, W] i32
    // d_in[2] = lesion_ids (assumed 1..Q per reference arange)
    float* ws  = (float*)d_ws;
    float* out = (float*)d_out;

    k_zero<<<(WS_TOTAL + 255) / 256, 256, 0, stream>>>(ws, WS_TOTAL);

    dim3 g1(256, NB);
    k_sums<<<g1, 256, 0, stream>>>(ft, tm, ws);

    k_norm<<<1, NB * BQ * CCH, 0, stream>>>(ws);

    dim3 g3(NVOX / (16 * 8), NB);              // 8 waves/block, 16 cols/wave
    k_loss<<<g3, 256, 0, stream>>>(ft, tm, ws + WS_TN, ws + WS_ACC);

    k_final<<<1, 1, 0, stream>>>(ws, out);
}